// OnlineHadamard_20710332301870
// MI455X (gfx1250) — compile-verified
//
#include <hip/hip_runtime.h>

// ext-vector types matching gfx1250 WMMA f32 operand register counts (wave32):
//   A/B 16x4 / 4x16 fp32 -> 2 VGPRs/lane ; C/D 16x16 fp32 -> 8 VGPRs/lane
typedef __attribute__((ext_vector_type(2))) float v2f;
typedef __attribute__((ext_vector_type(8))) float v8f;

// Hadamard entry H16[a][b] = +1 / -1 by parity of popcount(a & b). H is symmetric.
__device__ __forceinline__ float hadsign(int a, int b) {
    return (__builtin_popcount(a & b) & 1) ? -1.0f : 1.0f;
}

// One 16x16x16 product  D = A * (H16*scale)  as 4 chained v_wmma_f32_16x16x4_f32.
// A read from LDS at `base` with element strides (a_rs, a_cs) for (row M, col K);
// D written with strides (d_rs, d_cs) for (row M, col N).  B (the Hadamard signs)
// is precomputed per-lane by the caller: bsg[ch] holds the two B values this lane
// contributes for K-chunk ch.  H = H^T makes B's (K,N) lane-layout orientation moot.
//
// VGPR layouts (ISA 7.12.2, wave32):
//   A (16x4 f32):  lane l -> M = l&15 ; K = 2*(l>>4) + vgpr   (vgpr 0..1)
//   B (4x16 f32):  lane l -> N = l&15 ; K = 2*(l>>4) + vgpr   (half-split mirror of C/D)
//   C/D (16x16):   vgpr v, lane l -> M = v + 8*(l>>4), N = l&15
__device__ __forceinline__ void fwht16_wmma(float* lds, int base,
                                            int a_rs, int a_cs,
                                            int d_rs, int d_cs,
                                            const v2f bsg[4]) {
    const int lane = threadIdx.x & 31;
    const int m    = lane & 15;
    const int half = lane >> 4;   // 0 or 1
    v8f c = {};
#pragma unroll
    for (int ch = 0; ch < 4; ++ch) {          // K chunks of 4
        const int k0 = 4 * ch + 2 * half;
        v2f a;
        a.x = lds[base + m * a_rs + (k0 + 0) * a_cs];
        a.y = lds[base + m * a_rs + (k0 + 1) * a_cs];
        // 8 args: (neg_a, A, neg_b, B, c_mod, C, reuse_a, reuse_b); f32 NEG must be 0
        c = __builtin_amdgcn_wmma_f32_16x16x4_f32(false, a, false, bsg[ch], (short)0, c,
                                                  false, false);
    }
#pragma unroll
    for (int v = 0; v < 8; ++v)
        lds[base + (v + 8 * half) * d_rs + m * d_cs] = c[v];
}

// One workgroup (256 threads = 8 wave32) per row of 4096 fp32.
// Row viewed as x[i][j][k], i,j,k in [0,16); linear index = i*256 + j*16 + k.
// FWHT_4096 = (H16 along k) then (H16 along j) then (H16 along i), scale 1/sqrt(32).
__global__ void __launch_bounds__(256)
fwht4096_wmma_kernel(const float* __restrict__ x, float* __restrict__ out) {
    __shared__ float lds[4096];                       // 16 KB of 320 KB/WGP
    const size_t row = blockIdx.x;
    const float* src = x + row * 4096;
    float*       dst = out + row * 4096;
    const int t = threadIdx.x;

    // ---- Async row load: global -> LDS directly (no VGPR staging), ASYNCcnt path.
    // SADDR form per ISA 15.18.3: vdst = LDS byte offset (relative to wave LDS_BASE),
    // vaddr = 32-bit byte offset, s[..] = 64-bit base.  4 x 16 B per thread = 16 KB.
#pragma unroll
    for (int q = 0; q < 4; ++q) {
        const unsigned off = (unsigned)(t * 16 + q * 4096);   // bytes
        asm volatile("global_load_async_to_lds_b128 %0, %1, %2 offset:0"
                     :: "v"(off), "v"(off), "s"(src) : "memory");
    }
    asm volatile("s_wait_asynccnt 0x0" ::: "memory");
    __syncthreads();

    // ---- Precompute per-lane Hadamard B operands once (shared by every tile/stage).
    const int lane = t & 31;
    const int m    = lane & 15;
    const int half = lane >> 4;
    const float SCALE = 0.17677669529663687f;         // 1/sqrt(32), matches reference
    v2f bs1[4], bs3[4];
#pragma unroll
    for (int ch = 0; ch < 4; ++ch) {
        const int k0 = 4 * ch + 2 * half;
        bs1[ch].x = hadsign(k0 + 0, m);
        bs1[ch].y = hadsign(k0 + 1, m);
        bs3[ch].x = bs1[ch].x * SCALE;                // fold final scale into stage 3's H
        bs3[ch].y = bs1[ch].y * SCALE;
    }

    const int wave = t >> 5;                          // 8 waves, 2 tiles each per stage

    // Stage 1 (along k): tile i, M[j][k] at base i*256, rs=16, cs=1.  D = M*H, same strides.
#pragma unroll
    for (int s = 0; s < 2; ++s) {
        const int i = 2 * wave + s;
        fwht16_wmma(lds, i * 256, /*a*/16, 1, /*d*/16, 1, bs1);
    }
    __syncthreads();

    // Stage 2 (along j): out = H*M = (M^T * H)^T -> A=M^T (rs=1, cs=16), D transposed.
#pragma unroll
    for (int s = 0; s < 2; ++s) {
        const int i = 2 * wave + s;
        fwht16_wmma(lds, i * 256, /*a*/1, 16, /*d*/1, 16, bs1);
    }
    __syncthreads();

    // Stage 3 (along i): tile j, P[i][k] at base j*16, i-stride 256.
    // out = H*P = (P^T * H)^T -> A rs=1 (k), cs=256 (i); D rs=1, cs=256; scaled H.
#pragma unroll
    for (int s = 0; s < 2; ++s) {
        const int j = 2 * wave + s;
        fwht16_wmma(lds, j * 16, /*a*/1, 256, /*d*/1, 256, bs3);
    }
    __syncthreads();                                  // stage-3 DS writes visible to all

    // ---- Async row store: LDS -> global (VADDR, VSRC_lds, saddr).  The implicit
    // wait-idle at S_ENDPGM drains ASYNCcnt before the wave retires.
#pragma unroll
    for (int q = 0; q < 4; ++q) {
        const unsigned off = (unsigned)(t * 16 + q * 4096);   // bytes
        asm volatile("global_store_async_from_lds_b128 %0, %1, %2 offset:0"
                     :: "v"(off), "v"(off), "s"(dst) : "memory");
    }
}

extern "C" void kernel_launch(void* const* d_in, const int* in_sizes, int n_in,
                              void* d_out, int out_size, void* d_ws, size_t ws_size,
                              hipStream_t stream) {
    const float* x  = (const float*)d_in[0];
    float* out      = (float*)d_out;
    const int rows  = in_sizes[0] / 4096;             // 8192 for the reference shapes
    fwht4096_wmma_kernel<<<dim3(rows), dim3(256), 0, stream>>>(x, out);
}